// GConvLSTMCore_71923522339512
// MI455X (gfx1250) — compile-verified
//
#include <hip/hip_runtime.h>
#include <hip/hip_bf16.h>
#include <math.h>

// ---------------------------------------------------------------------------
// GConvLSTM core for MI455X (gfx1250, wave32, WMMA f32 16x16x4).
// B=4, N=4096, CI=CO=32, K=3 Chebyshev.
// ---------------------------------------------------------------------------

typedef float v2f __attribute__((ext_vector_type(2)));
typedef float v8f __attribute__((ext_vector_type(8)));

static constexpr int NB  = 4;
static constexpr int NN  = 4096;
static constexpr int F   = 32;
static constexpr int BN  = NB * NN;        // 16384 rows total
static constexpr int KF  = 192;            // concat feature width (6 * 32)
static constexpr int GC  = 128;            // gate output cols (4 gates * 32)

// workspace layout (in floats)
static constexpr size_t OFF_WT   = 0;                       // 192*128 = 24576
static constexpr size_t OFF_BIAS = OFF_WT + (size_t)KF*GC;  // 128
static constexpr size_t OFF_T1X  = 24704;                   // BN*F
static constexpr size_t OFF_T1H  = OFF_T1X + (size_t)BN*F;
static constexpr size_t OFF_T2X  = OFF_T1H + (size_t)BN*F;
static constexpr size_t OFF_T2H  = OFF_T2X + (size_t)BN*F;

// --- CDNA5 async global->LDS path (ASYNCcnt), with sync fallback ------------
#if defined(__has_builtin)
#  if __has_builtin(__builtin_amdgcn_global_load_async_to_lds_b128)
#    define USE_ASYNC_LDS 1
#  endif
#endif
#ifndef USE_ASYNC_LDS
#  define USE_ASYNC_LDS 0
#endif

#if USE_ASYNC_LDS
// builtin expects: (b128 vector ptr in AS1, b128 vector ptr in AS3, imm, imm)
typedef int b128_t __attribute__((vector_size(16)));
typedef __attribute__((address_space(1))) b128_t GlobalB128;
typedef __attribute__((address_space(3))) b128_t LdsB128;
__device__ __forceinline__ void async_b128(const float* gsrc, float* ldst) {
    __builtin_amdgcn_global_load_async_to_lds_b128(
        (GlobalB128*)(gsrc), (LdsB128*)(ldst), 0, 0);
}
#endif

__device__ __forceinline__ void wait_async0() {
#if USE_ASYNC_LDS
#  if __has_builtin(__builtin_amdgcn_s_wait_asynccnt)
    __builtin_amdgcn_s_wait_asynccnt(0);
#  else
    asm volatile("s_wait_asynccnt 0x0" ::: "memory");
#  endif
#endif
}

// ---------------------------------------------------------------------------
// Kernel 1: pack gate weights into Wt[col=128][k=192] and fuse bias vectors.
// feature index f = src*96 + k*32 + ci  (src: 0=X,1=H);  col = gate*32 + co.
// ---------------------------------------------------------------------------
__global__ __launch_bounds__(256)
void pack_weights(const float* __restrict__ Wxi, const float* __restrict__ Whi,
                  const float* __restrict__ Wxf, const float* __restrict__ Whf,
                  const float* __restrict__ Wxc, const float* __restrict__ Whc,
                  const float* __restrict__ Wxo, const float* __restrict__ Who,
                  const float* __restrict__ bxi, const float* __restrict__ bhi,
                  const float* __restrict__ b_i,
                  const float* __restrict__ bxf, const float* __restrict__ bhf,
                  const float* __restrict__ b_f,
                  const float* __restrict__ bxc, const float* __restrict__ bhc,
                  const float* __restrict__ b_c,
                  const float* __restrict__ bxo, const float* __restrict__ bho,
                  const float* __restrict__ b_o,
                  float* __restrict__ Wt, float* __restrict__ bias)
{
    int t = blockIdx.x * 256 + threadIdx.x;
    if (t < GC) {
        int g = t >> 5, co = t & 31;
        const float *ba, *bb, *bc;
        switch (g) {
            case 0:  ba = bxi; bb = bhi; bc = b_i; break;
            case 1:  ba = bxf; bb = bhf; bc = b_f; break;
            case 2:  ba = bxc; bb = bhc; bc = b_c; break;
            default: ba = bxo; bb = bho; bc = b_o; break;
        }
        bias[t] = ba[co] + bb[co] + bc[co];
    }
    if (t < KF * GC) {
        int col = t / KF, f = t % KF;
        int g = col >> 5, co = col & 31;
        int src = f / 96, rem = f % 96;
        int k = rem >> 5, ci = rem & 31;
        const float* W;
        if (src == 0) {
            switch (g) { case 0: W = Wxi; break; case 1: W = Wxf; break;
                         case 2: W = Wxc; break; default: W = Wxo; break; }
        } else {
            switch (g) { case 0: W = Whi; break; case 1: W = Whf; break;
                         case 2: W = Whc; break; default: W = Who; break; }
        }
        Wt[t] = W[(k * 32 + ci) * 32 + co];   // W is [K][ci][co] row-major
    }
}

// ---------------------------------------------------------------------------
// Kernel 2: Y[b] = alpha * (L @ R[b]) + beta * P[b], for 8 sources
// (grid.y = batch*2 + {X,H}).  Block: 128 threads = 4 waves; each wave owns a
// 32-row x 32-col strip of a 128x32 tile (4 WMMA accumulators -> 1 LDS
// fragment load per WMMA).  K staged through double-buffered LDS, 32/chunk,
// via async global->LDS DMA when available.
// ---------------------------------------------------------------------------
__global__ __launch_bounds__(128)
void cheb_gemm(const float* __restrict__ Lm,
               const float* __restrict__ RX, const float* __restrict__ RH,
               const float* __restrict__ PX, const float* __restrict__ PH,
               float* __restrict__ YX, float* __restrict__ YH,
               float alpha, float beta)
{
    __shared__ float Ls[2][128 * 36];   // L tile, padded stride 36
    __shared__ float Rs[2][32 * 36];    // R tile [k][c], padded stride 36

    const int tid  = threadIdx.x;
    const int wave = tid >> 5;          // 0..3
    const int lane = tid & 31;
    const int lm   = lane & 15;         // M for A-frag, N for B/C-frag
    const int hi   = lane >> 4;         // K-split for A/B, M-split for C
    const int rtile = blockIdx.x * 128;
    const int s    = blockIdx.y;        // 0..7
    const int b    = s >> 1;
    const int isH  = s & 1;

    const float* R = (isH ? RH : RX) + (size_t)b * NN * F;
    const float* P = (isH ? PH : PX);
    if (P) P += (size_t)b * NN * F;
    float* Y = (isH ? YH : YX) + (size_t)b * NN * F;

    v8f c00 = {}, c01 = {}, c10 = {}, c11 = {};
    const int rb = wave * 32;

    // stage one K-slab (L[rtile:+128, kb:+32] and R[kb:+32, 0:32]) into buf
    auto stage = [&](int buf, int kb) {
#if USE_ASYNC_LDS
        #pragma unroll
        for (int j = 0; j < 8; ++j) {
            int q = tid + 128 * j;            // 0..1023 float4 slots
            int row = q >> 3, seg = q & 7;
            async_b128(Lm + (size_t)(rtile + row) * NN + kb + seg * 4,
                       &Ls[buf][row * 36 + seg * 4]);
        }
        #pragma unroll
        for (int j = 0; j < 2; ++j) {
            int q = tid + 128 * j;            // 0..255 float4 slots
            int k = q >> 3, seg = q & 7;
            async_b128(R + (size_t)(kb + k) * F + seg * 4,
                       &Rs[buf][k * 36 + seg * 4]);
        }
#else
        #pragma unroll
        for (int j = 0; j < 8; ++j) {
            int q = tid + 128 * j;
            int row = q >> 3, seg = q & 7;
            float4 v = *(const float4*)(Lm + (size_t)(rtile + row) * NN + kb + seg * 4);
            *(float4*)(&Ls[buf][row * 36 + seg * 4]) = v;
        }
        #pragma unroll
        for (int j = 0; j < 2; ++j) {
            int q = tid + 128 * j;
            int k = q >> 3, seg = q & 7;
            float4 v = *(const float4*)(R + (size_t)(kb + k) * F + seg * 4);
            *(float4*)(&Rs[buf][k * 36 + seg * 4]) = v;
        }
#endif
    };

    stage(0, 0);
    wait_async0();
    __syncthreads();

    int cur = 0;
    for (int kb = 0; kb < NN; kb += 32) {
        if (kb + 32 < NN) stage(cur ^ 1, kb + 32);   // DMA next slab

        const float* Lsb = &Ls[cur][0];
        const float* Rsb = &Rs[cur][0];
        #pragma unroll
        for (int kk = 0; kk < 32; kk += 4) {
            int klo = kk + 2 * hi;                   // each lane: 2 consecutive K
            v2f a0 = *(const v2f*)(Lsb + (rb + lm) * 36 + klo);
            v2f a1 = *(const v2f*)(Lsb + (rb + 16 + lm) * 36 + klo);
            v2f b0 = { Rsb[klo * 36 + lm],      Rsb[(klo + 1) * 36 + lm] };
            v2f b1 = { Rsb[klo * 36 + lm + 16], Rsb[(klo + 1) * 36 + lm + 16] };
            c00 = __builtin_amdgcn_wmma_f32_16x16x4_f32(false, a0, false, b0,
                                                        (short)0, c00, false, false);
            c01 = __builtin_amdgcn_wmma_f32_16x16x4_f32(false, a0, false, b1,
                                                        (short)0, c01, false, false);
            c10 = __builtin_amdgcn_wmma_f32_16x16x4_f32(false, a1, false, b0,
                                                        (short)0, c10, false, false);
            c11 = __builtin_amdgcn_wmma_f32_16x16x4_f32(false, a1, false, b1,
                                                        (short)0, c11, false, false);
        }
        wait_async0();      // next slab resident (this wave's DMAs)
        __syncthreads();    // all waves done computing + DMAing
        cur ^= 1;
    }

    // epilogue: Y = alpha*acc + beta*P   (C/D layout: vgpr r -> M = r + 8*hi)
    #pragma unroll
    for (int r = 0; r < 8; ++r) {
        int row0 = rtile + rb + hi * 8 + r;
        int row1 = row0 + 16;
        size_t b0o = (size_t)row0 * F;
        size_t b1o = (size_t)row1 * F;
        float y00 = alpha * c00[r], y01 = alpha * c01[r];
        float y10 = alpha * c10[r], y11 = alpha * c11[r];
        if (beta != 0.0f) {
            y00 += beta * P[b0o + lm];      y01 += beta * P[b0o + lm + 16];
            y10 += beta * P[b1o + lm];      y11 += beta * P[b1o + lm + 16];
        }
        Y[b0o + lm]      = y00;
        Y[b0o + lm + 16] = y01;
        Y[b1o + lm]      = y10;
        Y[b1o + lm + 16] = y11;
    }
}

// ---------------------------------------------------------------------------
// Kernel 3: gates.  Per 64-row tile: Z[64][128] = Feat[64][192] @ Wt^T + bias
// via WMMA, then the LSTM nonlinearity.  Feature rows staged in LDS; Z aliases
// the same LDS after a barrier.
// ---------------------------------------------------------------------------
__global__ __launch_bounds__(256)
void gate_kernel(const float* __restrict__ X,   const float* __restrict__ H,
                 const float* __restrict__ T1X, const float* __restrict__ T1H,
                 const float* __restrict__ T2X, const float* __restrict__ T2H,
                 const float* __restrict__ C,
                 const float* __restrict__ Wt,   // [128][192]
                 const float* __restrict__ bias, // [128]
                 const float* __restrict__ wci, const float* __restrict__ wcf,
                 const float* __restrict__ wco,
                 float* __restrict__ Hout, float* __restrict__ Cout)
{
    __shared__ float smem[64 * 196];   // phase 1: Feat[64][196]; phase 2: Z[64][132]

    const int tid  = threadIdx.x;
    const int wave = tid >> 5;
    const int lane = tid & 31;
    const int lm   = lane & 15;
    const int hi   = lane >> 4;
    const int rtile = blockIdx.x * 64;  // flattened b*N+n row base

    // stage features: row*192 = [x | t1x | t2x | h | t1h | t2h]
    for (int idx = tid; idx < 64 * 48; idx += 256) {
        int row = idx / 48, seg = idx % 48;
        int sidx = seg >> 3, off = (seg & 7) * 4;
        const float* sp;
        switch (sidx) {
            case 0:  sp = X;   break;
            case 1:  sp = T1X; break;
            case 2:  sp = T2X; break;
            case 3:  sp = H;   break;
            case 4:  sp = T1H; break;
            default: sp = T2H; break;
        }
        float4 v = *(const float4*)(sp + (size_t)(rtile + row) * 32 + off);
        *(float4*)(smem + row * 196 + sidx * 32 + off) = v;
    }
    __syncthreads();

    // WMMA: wave -> row block (wave&3)*16, col half (wave>>2)*64 (4 col blocks)
    const int rb    = (wave & 3) * 16;
    const int cbase = (wave >> 2) * 64;
    v8f acc[4] = {};
    for (int kk = 0; kk < KF; kk += 4) {
        int klo = kk + 2 * hi;
        v2f a = *(const v2f*)(smem + (rb + lm) * 196 + klo);
        #pragma unroll
        for (int cb = 0; cb < 4; ++cb) {
            int col = cbase + cb * 16 + lm;
            v2f bfrag = *(const v2f*)(Wt + (size_t)col * KF + klo);
            acc[cb] = __builtin_amdgcn_wmma_f32_16x16x4_f32(false, a, false, bfrag,
                                                            (short)0, acc[cb],
                                                            false, false);
        }
    }
    __syncthreads();

    // spill Z (+bias) into aliased LDS: Z[64][132]
    #pragma unroll
    for (int cb = 0; cb < 4; ++cb) {
        int col = cbase + cb * 16 + lm;
        float bv = bias[col];
        #pragma unroll
        for (int r = 0; r < 8; ++r) {
            int row = rb + hi * 8 + r;
            smem[row * 132 + col] = acc[cb][r] + bv;
        }
    }
    __syncthreads();

    // elementwise LSTM gates
    for (int e = tid; e < 64 * 32; e += 256) {
        int row = e >> 5, co = e & 31;
        int gr = rtile + row;
        float zi = smem[row * 132 +       co];
        float zf = smem[row * 132 +  32 + co];
        float zc = smem[row * 132 +  64 + co];
        float zo = smem[row * 132 +  96 + co];
        float cv = C[(size_t)gr * 32 + co];
        float Ig = 1.0f / (1.0f + expf(-(zi + wci[co] * cv)));
        float Fg = 1.0f / (1.0f + expf(-(zf + wcf[co] * cv)));
        float Tg = tanhf(zc);
        float cn = Fg * cv + Ig * Tg;
        float Og = 1.0f / (1.0f + expf(-(zo + wco[co] * cn)));
        Hout[(size_t)gr * 32 + co] = Og * tanhf(cn);
        Cout[(size_t)gr * 32 + co] = cn;
    }
}

// ---------------------------------------------------------------------------
extern "C" void kernel_launch(void* const* d_in, const int* in_sizes, int n_in,
                              void* d_out, int out_size, void* d_ws, size_t ws_size,
                              hipStream_t stream)
{
    const float* X   = (const float*)d_in[0];
    const float* Lm  = (const float*)d_in[1];
    const float* H   = (const float*)d_in[2];
    const float* C   = (const float*)d_in[3];
    const float* Wxi = (const float*)d_in[4];
    const float* bxi = (const float*)d_in[5];
    const float* Whi = (const float*)d_in[6];
    const float* bhi = (const float*)d_in[7];
    const float* Wxf = (const float*)d_in[8];
    const float* bxf = (const float*)d_in[9];
    const float* Whf = (const float*)d_in[10];
    const float* bhf = (const float*)d_in[11];
    const float* Wxc = (const float*)d_in[12];
    const float* bxc = (const float*)d_in[13];
    const float* Whc = (const float*)d_in[14];
    const float* bhc = (const float*)d_in[15];
    const float* Wxo = (const float*)d_in[16];
    const float* bxo = (const float*)d_in[17];
    const float* Who = (const float*)d_in[18];
    const float* bho = (const float*)d_in[19];
    const float* wci = (const float*)d_in[20];
    const float* wcf = (const float*)d_in[21];
    const float* wco = (const float*)d_in[22];
    const float* b_i = (const float*)d_in[23];
    const float* b_f = (const float*)d_in[24];
    const float* b_c = (const float*)d_in[25];
    const float* b_o = (const float*)d_in[26];

    float* ws   = (float*)d_ws;
    float* Wt   = ws + OFF_WT;
    float* bias = ws + OFF_BIAS;
    float* T1X  = ws + OFF_T1X;
    float* T1H  = ws + OFF_T1H;
    float* T2X  = ws + OFF_T2X;
    float* T2H  = ws + OFF_T2H;

    float* Hout = (float*)d_out;
    float* Cout = Hout + (size_t)BN * F;

    pack_weights<<<(KF * GC + 255) / 256, 256, 0, stream>>>(
        Wxi, Whi, Wxf, Whf, Wxc, Whc, Wxo, Who,
        bxi, bhi, b_i, bxf, bhf, b_f, bxc, bhc, b_c, bxo, bho, b_o,
        Wt, bias);

    dim3 g1(NN / 128, 2 * NB);
    // T1 = L @ {X,H}
    cheb_gemm<<<g1, 128, 0, stream>>>(Lm, X, H, nullptr, nullptr,
                                      T1X, T1H, 1.0f, 0.0f);
    // T2 = 2 * L @ T1 - {X,H}
    cheb_gemm<<<g1, 128, 0, stream>>>(Lm, T1X, T1H, X, H,
                                      T2X, T2H, 2.0f, -1.0f);

    gate_kernel<<<BN / 64, 256, 0, stream>>>(X, H, T1X, T1H, T2X, T2H, C,
                                             Wt, bias, wci, wcf, wco,
                                             Hout, Cout);
}